// GroupLinear_45277545234494
// MI455X (gfx1250) — compile-verified
//
#include <hip/hip_runtime.h>

typedef float v2f __attribute__((ext_vector_type(2)));
typedef float v8f __attribute__((ext_vector_type(8)));

#define IN_CH      4096
#define OUT_CH     4096
#define GSIZE      64      // per-group in/out width (IN_SCALE == OUT_SCALE == 64)
#define GROUPS     64
#define WAVES_PB   8       // 256 threads / wave32

// y[t, g*64 + o] = sum_i x[t, g*64 + i] * W[(g*64+o)*IN_CH + g*64 + i]
// Per group: GEMM (T x 64) * (64 x 64)^T using V_WMMA_F32_16X16X4_F32.
__global__ __launch_bounds__(256) void group_linear_wmma(
    const float* __restrict__ x, const float* __restrict__ W,
    float* __restrict__ y, int T, int nTiles)
{
    __shared__ float wblk[GSIZE * GSIZE];   // 16 KB: Wdiag[g] row-major [o][i]

    const int g   = blockIdx.y;
    const int tid = threadIdx.x;

    // Stage this group's diagonal 64x64 weight block into LDS (coalesced rows).
    const float* Wg = W + (size_t)g * GSIZE * IN_CH + (size_t)g * GSIZE;
    for (int idx = tid; idx < GSIZE * GSIZE; idx += 256) {
        int o = idx >> 6;
        int i = idx & 63;
        wblk[idx] = Wg[(size_t)o * IN_CH + i];
    }
    __syncthreads();

    const int lane  = tid & 31;
    const int wave  = tid >> 5;
    const int lm    = lane & 15;            // M (tokens) / N (outputs) index in fragment
    const int khalf = (lane >> 4) << 1;     // K offset of .x element: 0 (lanes 0-15) or 2

    // B fragments (4x16, K-major per ISA layout): lane holds B[k0+khalf][n], B[k0+khalf+1][n]
    // B[k=i][n=o] = Wdiag[o][i] = wblk[o*64 + i]. Keep all 64 fragments resident.
    v2f bfrag[16][4];
    #pragma unroll
    for (int kb = 0; kb < 16; ++kb) {
        #pragma unroll
        for (int nt = 0; nt < 4; ++nt) {
            const float* p = &wblk[(nt * 16 + lm) * GSIZE + kb * 4 + khalf];
            bfrag[kb][nt].x = p[0];
            bfrag[kb][nt].y = p[1];
        }
    }

    // Stream 16-token tiles; each wave owns the full 64-wide output slab of its tile,
    // so x traffic from HBM is exactly once.
    for (int tile = blockIdx.x * WAVES_PB + wave; tile < nTiles;
         tile += gridDim.x * WAVES_PB) {
        const int t0 = tile * 16;
        const float* xrow = x + (size_t)(t0 + lm) * IN_CH + g * GSIZE;

        v8f acc0 = {}, acc1 = {}, acc2 = {}, acc3 = {};

        #pragma unroll
        for (int kb = 0; kb < 16; ++kb) {
            // A fragment 16x4: lane holds x[t0+lm][k0+khalf], x[t0+lm][k0+khalf+1]
            v2f a = *(const v2f*)(xrow + kb * 4 + khalf);
            acc0 = __builtin_amdgcn_wmma_f32_16x16x4_f32(
                false, a, false, bfrag[kb][0], (short)0, acc0, false, false);
            acc1 = __builtin_amdgcn_wmma_f32_16x16x4_f32(
                false, a, false, bfrag[kb][1], (short)0, acc1, false, false);
            acc2 = __builtin_amdgcn_wmma_f32_16x16x4_f32(
                false, a, false, bfrag[kb][2], (short)0, acc2, false, false);
            acc3 = __builtin_amdgcn_wmma_f32_16x16x4_f32(
                false, a, false, bfrag[kb][3], (short)0, acc3, false, false);
        }

        // D layout: VGPR v, lanes 0-15 -> M=v, lanes 16-31 -> M=v+8; N = lane&15.
        const int mhalf = (lane >> 4) << 3;     // 0 or 8
        float* yrow = y + (size_t)(t0 + mhalf) * OUT_CH + g * GSIZE + lm;
        #pragma unroll
        for (int v = 0; v < 8; ++v) {
            yrow[(size_t)v * OUT_CH +  0] = acc0[v];
            yrow[(size_t)v * OUT_CH + 16] = acc1[v];
            yrow[(size_t)v * OUT_CH + 32] = acc2[v];
            yrow[(size_t)v * OUT_CH + 48] = acc3[v];
        }
    }
}

extern "C" void kernel_launch(void* const* d_in, const int* in_sizes, int n_in,
                              void* d_out, int out_size, void* d_ws, size_t ws_size,
                              hipStream_t stream) {
    const float* x = (const float*)d_in[0];
    const float* W = (const float*)d_in[1];
    float* y = (float*)d_out;

    const int T      = in_sizes[0] / IN_CH;   // 8 * 2048 = 16384 tokens
    const int nTiles = (T + 15) / 16;         // 1024 tiles of 16 tokens

    dim3 grid(32, GROUPS);                    // 2048 workgroups, 8 waves each
    group_linear_wmma<<<grid, 256, 0, stream>>>(x, W, y, T, nTiles);
}